// NGP_30004641530342
// MI455X (gfx1250) — compile-verified
//
#include <hip/hip_runtime.h>
#include <hip/hip_bf16.h>

typedef __attribute__((ext_vector_type(2))) float v2f;
typedef __attribute__((ext_vector_type(8))) float v8f;

#define WPB 4                 // waves per block (wave32)
#define TLOG 19
#define TMASK ((1u << TLOG) - 1u)

// floor(16 * (128^(1/15))^k), k=0..15 (matches reference _resolutions())
__device__ __constant__ float c_res[16] = {
    16.f, 22.f, 30.f, 42.f, 58.f, 80.f, 111.f, 153.f,
    212.f, 294.f, 406.f, 561.f, 776.f, 1072.f, 1482.f, 2048.f};

// LDS layout (float offsets); row strides padded (34 / 66) to avoid bank conflicts
#define OFF_W0X 0                      // xyz_w0  64 x 32 -> stride 34 (2176)
#define OFF_WOX 2176                   // xyz_wout 16 x 64 -> stride 66 (1056)
#define OFF_W0R 3232                   // rgb_w0  64 x 32 -> stride 34 (2176)
#define OFF_W1R 5408                   // rgb_w1  64 x 64 -> stride 66 (4224)
#define OFF_WOR 9632                   // rgb_wout padded 16 x 64 -> stride 66 (1056)
#define OFF_BUF 10688                  // per-wave activation buffer 16 x 66
#define BUF_STRIDE 1056
#define SMEM_FLOATS (OFF_BUF + WPB * BUF_STRIDE)   // 14912 floats = 59648 B

__device__ __forceinline__ v8f wmma4(v2f a, v2f b, v8f c) {
  // V_WMMA_F32_16X16X4_F32 : D = A(16x4) x B(4x16) + C(16x16), all fp32
  return __builtin_amdgcn_wmma_f32_16x16x4_f32(
      false, a, false, b, (short)0, c, false, false);
}

// Intra-wave LDS ordering: LDS is in-order per wave (DScnt); the hardware wait
// plus the "memory" clobber (compiler scheduling fence) makes prior ds_stores
// visible to subsequent ds_loads of the same wave. No block barrier needed for
// the per-wave scratch buffer.
__device__ __forceinline__ void wave_lds_sync() {
  asm volatile("s_wait_dscnt 0x0" ::: "memory");
}

__global__ __launch_bounds__(128) void ngp_fwd(
    const float* __restrict__ xin, const float* __restrict__ din,
    const float* __restrict__ table,
    const float* __restrict__ xyz_w0, const float* __restrict__ xyz_wout,
    const float* __restrict__ rgb_w0, const float* __restrict__ rgb_w1,
    const float* __restrict__ rgb_wout,
    float* __restrict__ sig, float* __restrict__ rgb, int n) {
  __shared__ __align__(16) float smem[SMEM_FLOATS];
  const int tid = threadIdx.x;

  // ---- stage weights into LDS (once per block) ----
  for (int i = tid; i < 64 * 32; i += 128)
    smem[OFF_W0X + (i >> 5) * 34 + (i & 31)] = xyz_w0[i];
  for (int i = tid; i < 16 * 64; i += 128)
    smem[OFF_WOX + (i >> 6) * 66 + (i & 63)] = xyz_wout[i];
  for (int i = tid; i < 64 * 32; i += 128)
    smem[OFF_W0R + (i >> 5) * 34 + (i & 31)] = rgb_w0[i];
  for (int i = tid; i < 64 * 64; i += 128)
    smem[OFF_W1R + (i >> 6) * 66 + (i & 63)] = rgb_w1[i];
  for (int i = tid; i < 16 * 66; i += 128) smem[OFF_WOR + i] = 0.f;
  __syncthreads();
  for (int i = tid; i < 3 * 64; i += 128)
    smem[OFF_WOR + (i >> 6) * 66 + (i & 63)] = rgb_wout[i];
  __syncthreads();   // weights ready; waves are fully independent from here on

  const int wave = tid >> 5;
  const int lane = tid & 31;
  const int lhalf = lane >> 4;   // 0/1
  const int lrow = lane & 15;    // m / n index inside 16-tile
  float* buf = smem + OFF_BUF + wave * BUF_STRIDE;   // 16 x 66 per-wave scratch

  const int numTiles = n >> 4;
  const int gwave = blockIdx.x * WPB + wave;
  const int nwaves = gridDim.x * WPB;

  for (int tile = gwave; tile < numTiles; tile += nwaves) {
    const int base = tile << 4;

    // ================= stage 1: multires hash encode -> buf[p][0..31] =====
    {
      const int p = lrow;
      const int g = base + p;
      const float xn0 = xin[g * 3 + 0] + 0.5f;
      const float xn1 = xin[g * 3 + 1] + 0.5f;
      const float xn2 = xin[g * 3 + 2] + 0.5f;
#pragma unroll
      for (int i = 0; i < 8; ++i) {
        const int lev = lhalf * 8 + i;
        const float r = c_res[lev];
        const float p0 = xn0 * r, p1 = xn1 * r, p2 = xn2 * r;
        const float f0 = floorf(p0), f1 = floorf(p1), f2 = floorf(p2);
        const float fr0 = p0 - f0, fr1 = p1 - f1, fr2 = p2 - f2;
        const unsigned xi0 = (unsigned)f0, xi1 = (unsigned)f1,
                       xi2 = (unsigned)f2;
        float a0 = 0.f, a1 = 0.f;
#pragma unroll
        for (int c = 0; c < 8; ++c) {
          const unsigned cx = xi0 + ((c >> 2) & 1);
          const unsigned cy = xi1 + ((c >> 1) & 1);
          const unsigned cz = xi2 + (c & 1);
          const unsigned h =
              (cx ^ (cy * 2654435761u) ^ (cz * 805459861u)) & TMASK;
          const v2f ft =
              *(const v2f*)(table + (((size_t)lev << TLOG) + h) * 2);
          const float w = ((c & 4) ? fr0 : 1.f - fr0) *
                          ((c & 2) ? fr1 : 1.f - fr1) *
                          ((c & 1) ? fr2 : 1.f - fr2);
          a0 += w * ft.x;
          a1 += w * ft.y;
        }
        buf[p * 66 + 2 * lev] = a0;
        buf[p * 66 + 2 * lev + 1] = a1;
      }
    }
    wave_lds_sync();

    // ========= stage 2: xyz layer1 (32->64) + ReLU, 4 parallel chains =====
    v8f acc[4] = {{}, {}, {}, {}};
#pragma unroll
    for (int ks = 0; ks < 8; ++ks) {
      v2f a = *(const v2f*)&buf[lrow * 66 + 4 * ks + 2 * lhalf];
#pragma unroll
      for (int nt = 0; nt < 4; ++nt) {
        v2f b = *(const v2f*)&smem[OFF_W0X + (nt * 16 + lrow) * 34 + 4 * ks +
                                   2 * lhalf];
        acc[nt] = wmma4(a, b, acc[nt]);
      }
    }
#pragma unroll
    for (int nt = 0; nt < 4; ++nt)
#pragma unroll
      for (int r = 0; r < 8; ++r) acc[nt][r] = fmaxf(acc[nt][r], 0.f);
    wave_lds_sync();   // all A-frag reads done before overwrite
#pragma unroll
    for (int nt = 0; nt < 4; ++nt)
#pragma unroll
      for (int r = 0; r < 8; ++r)
        buf[(r + 8 * lhalf) * 66 + nt * 16 + lrow] = acc[nt][r];
    wave_lds_sync();

    // ========= stage 3: xyz layer2 (64->16), two parity chains, sigma =====
    v8f h0 = {}, h1 = {};
#pragma unroll
    for (int ks = 0; ks < 8; ++ks) {
      v2f a0 = *(const v2f*)&buf[lrow * 66 + 8 * ks + 2 * lhalf];
      v2f b0 = *(const v2f*)&smem[OFF_WOX + lrow * 66 + 8 * ks + 2 * lhalf];
      h0 = wmma4(a0, b0, h0);
      v2f a1 = *(const v2f*)&buf[lrow * 66 + 8 * ks + 4 + 2 * lhalf];
      v2f b1 =
          *(const v2f*)&smem[OFF_WOX + lrow * 66 + 8 * ks + 4 + 2 * lhalf];
      h1 = wmma4(a1, b1, h1);
    }
    v8f hacc = h0 + h1;
    if (lrow == 0) {
#pragma unroll
      for (int r = 0; r < 8; ++r)
        sig[base + r + 8 * lhalf] = expf(hacc[r]);
    }
    wave_lds_sync();

    // ================= stage 4: f = [sh || h] into buf[p][0..31] ==========
#pragma unroll
    for (int r = 0; r < 8; ++r)
      buf[(r + 8 * lhalf) * 66 + 16 + lrow] = hacc[r];
    if (lhalf == 0) {
      const int gd = base + lrow;
      const float dx0 = din[gd * 3 + 0], dy0 = din[gd * 3 + 1],
                  dz0 = din[gd * 3 + 2];
      const float inv = 1.f / sqrtf(dx0 * dx0 + dy0 * dy0 + dz0 * dz0);
      const float dx = dx0 * inv, dy = dy0 * inv, dz = dz0 * inv;
      const float xx = dx * dx, yy = dy * dy, zz = dz * dz;
      const float xy = dx * dy, yz = dy * dz, xz = dx * dz;
      float sh[16];
      sh[0] = 0.28209479177387814f;
      sh[1] = -0.4886025119029199f * dy;
      sh[2] = 0.4886025119029199f * dz;
      sh[3] = -0.4886025119029199f * dx;
      sh[4] = 1.0925484305920792f * xy;
      sh[5] = -1.0925484305920792f * yz;
      sh[6] = 0.31539156525252005f * (3.f * zz - 1.f);
      sh[7] = -1.0925484305920792f * xz;
      sh[8] = 0.5462742152960396f * (xx - yy);
      sh[9] = -0.5900435899266435f * dy * (3.f * xx - yy);
      sh[10] = 2.890611442640554f * xy * dz;
      sh[11] = -0.4570457994644658f * dy * (4.f * zz - xx - yy);
      sh[12] = 0.3731763325901154f * dz * (2.f * zz - 3.f * xx - 3.f * yy);
      sh[13] = -0.4570457994644658f * dx * (4.f * zz - xx - yy);
      sh[14] = 1.445305721320277f * dz * (xx - yy);
      sh[15] = -0.5900435899266435f * dx * (xx - 3.f * yy);
#pragma unroll
      for (int c = 0; c < 16; ++c) buf[lrow * 66 + c] = sh[c];
    }
    wave_lds_sync();

    // ========= stage 5: rgb layer0 (32->64) + ReLU, 4 parallel chains =====
#pragma unroll
    for (int nt = 0; nt < 4; ++nt) acc[nt] = (v8f){};
#pragma unroll
    for (int ks = 0; ks < 8; ++ks) {
      v2f a = *(const v2f*)&buf[lrow * 66 + 4 * ks + 2 * lhalf];
#pragma unroll
      for (int nt = 0; nt < 4; ++nt) {
        v2f b = *(const v2f*)&smem[OFF_W0R + (nt * 16 + lrow) * 34 + 4 * ks +
                                   2 * lhalf];
        acc[nt] = wmma4(a, b, acc[nt]);
      }
    }
#pragma unroll
    for (int nt = 0; nt < 4; ++nt)
#pragma unroll
      for (int r = 0; r < 8; ++r) acc[nt][r] = fmaxf(acc[nt][r], 0.f);
    wave_lds_sync();
#pragma unroll
    for (int nt = 0; nt < 4; ++nt)
#pragma unroll
      for (int r = 0; r < 8; ++r)
        buf[(r + 8 * lhalf) * 66 + nt * 16 + lrow] = acc[nt][r];
    wave_lds_sync();

    // ========= stage 6: rgb layer1 (64->64) + ReLU, 4 parallel chains =====
#pragma unroll
    for (int nt = 0; nt < 4; ++nt) acc[nt] = (v8f){};
#pragma unroll
    for (int ks = 0; ks < 16; ++ks) {
      v2f a = *(const v2f*)&buf[lrow * 66 + 4 * ks + 2 * lhalf];
#pragma unroll
      for (int nt = 0; nt < 4; ++nt) {
        v2f b = *(const v2f*)&smem[OFF_W1R + (nt * 16 + lrow) * 66 + 4 * ks +
                                   2 * lhalf];
        acc[nt] = wmma4(a, b, acc[nt]);
      }
    }
#pragma unroll
    for (int nt = 0; nt < 4; ++nt)
#pragma unroll
      for (int r = 0; r < 8; ++r) acc[nt][r] = fmaxf(acc[nt][r], 0.f);
    wave_lds_sync();
#pragma unroll
    for (int nt = 0; nt < 4; ++nt)
#pragma unroll
      for (int r = 0; r < 8; ++r)
        buf[(r + 8 * lhalf) * 66 + nt * 16 + lrow] = acc[nt][r];
    wave_lds_sync();

    // ==== stage 7: rgb out (64->3 padded to 16), two chains + sigmoid =====
    v8f o0 = {}, o1 = {};
#pragma unroll
    for (int ks = 0; ks < 8; ++ks) {
      v2f a0 = *(const v2f*)&buf[lrow * 66 + 8 * ks + 2 * lhalf];
      v2f b0 = *(const v2f*)&smem[OFF_WOR + lrow * 66 + 8 * ks + 2 * lhalf];
      o0 = wmma4(a0, b0, o0);
      v2f a1 = *(const v2f*)&buf[lrow * 66 + 8 * ks + 4 + 2 * lhalf];
      v2f b1 =
          *(const v2f*)&smem[OFF_WOR + lrow * 66 + 8 * ks + 4 + 2 * lhalf];
      o1 = wmma4(a1, b1, o1);
    }
    v8f oacc = o0 + o1;
    if (lrow < 3) {
#pragma unroll
      for (int r = 0; r < 8; ++r) {
        const float v = oacc[r];
        rgb[(base + r + 8 * lhalf) * 3 + lrow] = 1.f / (1.f + expf(-v));
      }
    }
    wave_lds_sync();   // protect buf before next tile's stage 1 stores
  }
}

extern "C" void kernel_launch(void* const* d_in, const int* in_sizes, int n_in,
                              void* d_out, int out_size, void* d_ws,
                              size_t ws_size, hipStream_t stream) {
  (void)n_in; (void)out_size; (void)d_ws; (void)ws_size;
  const float* x = (const float*)d_in[0];
  const float* d = (const float*)d_in[1];
  const float* table = (const float*)d_in[2];
  const float* xyz_w0 = (const float*)d_in[3];
  const float* xyz_wout = (const float*)d_in[4];
  const float* rgb_w0 = (const float*)d_in[5];
  const float* rgb_w1 = (const float*)d_in[6];
  const float* rgb_wout = (const float*)d_in[7];
  const int n = in_sizes[0] / 3;
  float* sig = (float*)d_out;
  float* rgb = (float*)d_out + n;
  ngp_fwd<<<dim3(512), dim3(128), 0, stream>>>(x, d, table, xyz_w0, xyz_wout,
                                               rgb_w0, rgb_w1, rgb_wout, sig,
                                               rgb, n);
}